// PointConvDW_90323162235006
// MI455X (gfx1250) — compile-verified
//
#include <hip/hip_runtime.h>

// CDNA5 / gfx1250, wave32. Fully fp32 path using V_WMMA_F32_16X16X4_F32.

typedef float v2f __attribute__((ext_vector_type(2)));
typedef float v8f __attribute__((ext_vector_type(8)));

#define LEAKY 0.1f

// Problem sizes (fixed by reference setup_inputs)
#define B_   2
#define CIN  64
#define COUT 64
#define HGT  120
#define WID  400
#define HW   (HGT * WID)   // 48000
#define NPT  (60 * 200)    // 12000 query points
#define KNB  16            // neighbors per point

__device__ __forceinline__ v8f wmma_f32_k4(v2f a, v2f b, v8f c) {
  // 8 args: (neg_a, A, neg_b, B, c_mod, C, reuse_a, reuse_b)
  return __builtin_amdgcn_wmma_f32_16x16x4_f32(
      false, a, false, b, (short)0, c, false, false);
}

// ---------------------------------------------------------------------------
// Kernel A: feature MLP  f = leaky_relu(W @ x + b), stored TRANSPOSED as
// f_t[b][hw][64] so the neighbor gather in kernel B reads one 256B line.
// One wave computes a 16(pixel) x 16(channel) tile; K=64 via 16 WMMA k-steps.
//
// Assumed fp32 16x16x4 fragment layouts (ISA 7.12.2):
//   A 16x4 : lane L<16 -> row M=L,  VGPR0=K0, VGPR1=K1; lanes 16..31 -> K2,K3
//   B 4x16 : lane L<16 -> col N=L,  VGPR0=K0, VGPR1=K1; lanes 16..31 -> K2,K3
//   C/D    : lane L, VGPR r -> M = r + 8*(L>=16), N = L%16
// ---------------------------------------------------------------------------
__global__ __launch_bounds__(256)
void feat_mlp_kernel(const float* __restrict__ feat,   // [B, CIN, HW]
                     const float* __restrict__ w,      // [COUT, CIN]
                     const float* __restrict__ bias,   // [COUT]
                     float* __restrict__ ft)           // [B, HW, COUT]
{
  const int wid  = (blockIdx.x * blockDim.x + threadIdx.x) >> 5;
  const int lane = threadIdx.x & 31;
  const int half = lane >> 4;
  const int l16  = lane & 15;

  const int ntile = wid & 3;                 // 4 channel tiles
  const int hwt   = (wid >> 2) % (HW / 16);  // 3000 pixel tiles
  const int b     = wid / (4 * (HW / 16));

  const int hw0 = hwt * 16;
  const int nb  = ntile * 16;

  const float* fb = feat + (size_t)b * CIN * HW;

  v8f acc = {};
#pragma unroll
  for (int kk = 0; kk < CIN / 4; ++kk) {
    const int kb = kk * 4 + half * 2;
    v2f a, bf;
    a.x = fb[(size_t)(kb + 0) * HW + hw0 + l16];
    a.y = fb[(size_t)(kb + 1) * HW + hw0 + l16];
    const float* wp = w + (nb + l16) * CIN + kb;   // B[k][n] = W[n][k]
    bf.x = wp[0];
    bf.y = wp[1];
    acc = wmma_f32_k4(a, bf, acc);
  }

  const float bv = bias[nb + l16];
  float* outp = ft + ((size_t)b * HW + hw0) * COUT + nb + l16;
#pragma unroll
  for (int r = 0; r < 8; ++r) {
    float v = acc[r] + bv;
    v = (v >= 0.f) ? v : LEAKY * v;                // leaky_relu
    outp[(size_t)(r + 8 * half) * COUT] = v;
  }
}

// ---------------------------------------------------------------------------
// Kernel B: one wave per (batch, query point).
//  rows M = the point's 16 neighbors.
//  - layers 1-2 of weight-net scalar (lane-pair per neighbor, each half of
//    the wave produces h2[0..15] or h2[16..31])
//  - h2 through LDS (stride 36 -> conflict-free) into A fragments
//  - layer 3 (32->64) via fp32 WMMA: 4 column tiles x 8 k-steps
//  - gather f_t rows (contiguous 256B each), multiply in-fragment,
//    max-reduce over the 16 neighbor rows (8 regs + shfl_xor 16)
// ---------------------------------------------------------------------------
__global__ __launch_bounds__(256)
void pointconv_kernel(const float* __restrict__ xyz,   // [B, 3, HW]
                      const float* __restrict__ sxyz,  // [B, 3, NPT]
                      const int*   __restrict__ knn,   // [B, NPT, KNB]
                      const unsigned char* __restrict__ msk8, // [B, NPT, KNB]
                      const float* __restrict__ w1, const float* __restrict__ b1,
                      const float* __restrict__ w2, const float* __restrict__ b2,
                      const float* __restrict__ w3, const float* __restrict__ b3,
                      const float* __restrict__ ft,    // [B, HW, COUT]
                      float* __restrict__ out)         // [B, COUT, NPT]
{
  __shared__ float h2lds[8][16 * 36];  // 8 waves/block, 16 rows, stride 36
  __shared__ int   idxlds[8][16];

  const int wib  = threadIdx.x >> 5;
  const int lane = threadIdx.x & 31;
  const int half = lane >> 4;
  const int m    = lane & 15;          // neighbor index handled by this lane

  const int wid = blockIdx.x * 8 + wib;
  const int b   = wid / NPT;
  const int p   = wid % NPT;

  const size_t ib = ((size_t)(b * NPT + p)) * KNB + m;
  const int  idx = knn[ib];
  const bool mk  = msk8[ib] != 0;

  const float* xb = xyz  + (size_t)b * 3 * HW;
  const float* sb = sxyz + (size_t)b * 3 * NPT;

  // knn_offset (masked neighbor xyz -> 0, exactly as reference)
  const float off0 = (mk ? xb[0 * HW + idx] : 0.f) - sb[0 * NPT + p];
  const float off1 = (mk ? xb[1 * HW + idx] : 0.f) - sb[1 * NPT + p];
  const float off2 = (mk ? xb[2 * HW + idx] : 0.f) - sb[2 * NPT + p];

  // layer 1: 3 -> 8 (both halves compute all 8; cheap)
  float h1[8];
#pragma unroll
  for (int j = 0; j < 8; ++j) {
    float s = b1[j] + w1[j * 3 + 0] * off0 + w1[j * 3 + 1] * off1 +
              w1[j * 3 + 2] * off2;
    h1[j] = s > 0.f ? s : 0.f;
  }

  // layer 2: 8 -> 32, each half produces 16 outputs, straight into LDS
  const int jb = half * 16;
#pragma unroll
  for (int j = 0; j < 16; ++j) {
    const float* wr = w2 + (jb + j) * 8;
    float s = b2[jb + j];
#pragma unroll
    for (int i = 0; i < 8; ++i) s += wr[i] * h1[i];
    h2lds[wib][m * 36 + jb + j] = (s > 0.f ? s : 0.f);
  }
  if (half == 0) idxlds[wib][m] = mk ? idx : -1;  // -1 encodes "gather -> 0"
  __syncthreads();

  // A fragments: A[m][k] = h2 of neighbor m, K=32 in 8 steps of 4
  v2f afrag[8];
#pragma unroll
  for (int kk = 0; kk < 8; ++kk) {
    const int kb = kk * 4 + half * 2;
    afrag[kk].x = h2lds[wib][m * 36 + kb + 0];
    afrag[kk].y = h2lds[wib][m * 36 + kb + 1];
  }

  const float* ftb = ft + (size_t)b * HW * COUT;
  float* ob = out + (size_t)b * COUT * NPT;

  float pm[4];

#pragma unroll
  for (int nt = 0; nt < 4; ++nt) {
    const int nb = nt * 16;
    v8f acc = {};
#pragma unroll
    for (int kk = 0; kk < 8; ++kk) {
      const int kb = kk * 4 + half * 2;
      const float* wp = w3 + (nb + m) * 32 + kb;   // B[k][n] = W3[n][k]
      v2f bf;
      bf.x = wp[0];
      bf.y = wp[1];
      acc = wmma_f32_k4(afrag[kk], bf, acc);
    }
    const float bv = b3[nb + m];
    float pmax = -__builtin_inff();
#pragma unroll
    for (int r = 0; r < 8; ++r) {
      float wv = acc[r] + bv;
      wv = wv > 0.f ? wv : 0.f;                    // relu -> wgt[row][col]
      const int row = r + 8 * half;                // neighbor index of this reg
      const int j = idxlds[wib][row];
      const float fv = (j >= 0) ? ftb[(size_t)j * COUT + nb + m] : 0.f;
      pmax = fmaxf(pmax, fv * wv);
    }
    // combine the two lane-halves (rows 0..7 vs 8..15)
    pmax = fmaxf(pmax, __shfl_xor(pmax, 16, 32));
    pm[nt] = pmax;
  }

  // stores hoisted after all WMMAs (EXEC all-1s during matrix ops)
  if (half == 0) {
#pragma unroll
    for (int nt = 0; nt < 4; ++nt)
      ob[(size_t)(nt * 16 + m) * NPT + p] = pm[nt];
  }
}

// ---------------------------------------------------------------------------
extern "C" void kernel_launch(void* const* d_in, const int* in_sizes, int n_in,
                              void* d_out, int out_size, void* d_ws,
                              size_t ws_size, hipStream_t stream) {
  const float* xyz   = (const float*)d_in[0];
  const float* feat  = (const float*)d_in[1];
  const float* sxyz  = (const float*)d_in[2];
  const int*   knn   = (const int*)d_in[3];
  const unsigned char* mask = (const unsigned char*)d_in[4]; // jax bool = 1B
  const float* mlp_w = (const float*)d_in[5];
  const float* mlp_b = (const float*)d_in[6];
  const float* w1    = (const float*)d_in[7];
  const float* b1    = (const float*)d_in[8];
  const float* w2    = (const float*)d_in[9];
  const float* b2    = (const float*)d_in[10];
  const float* w3    = (const float*)d_in[11];
  const float* b3    = (const float*)d_in[12];

  float* out = (float*)d_out;
  float* ft  = (float*)d_ws;   // B*HW*COUT floats = 24.6 MB scratch

  // Kernel A: B * (HW/16) * (COUT/16) = 24000 waves -> 3000 blocks x 8 waves
  feat_mlp_kernel<<<3000, 256, 0, stream>>>(feat, mlp_w, mlp_b, ft);

  // Kernel B: B * NPT = 24000 waves -> 3000 blocks x 8 waves
  pointconv_kernel<<<3000, 256, 0, stream>>>(xyz, sxyz, knn, mask,
                                             w1, b1, w2, b2, w3, b3, ft, out);
}